// GNNDecoder_39779987095912
// MI455X (gfx1250) — compile-verified
//
#include <hip/hip_runtime.h>

typedef __attribute__((ext_vector_type(2))) float v2f;
typedef __attribute__((ext_vector_type(8))) float v8f;

#define KDIM 128   // K dimension is 128 for every layer

// ---------------------------------------------------------------- utility
__global__ __launch_bounds__(256) void k_zero_int(int* p, int n) {
    int i = blockIdx.x * blockDim.x + threadIdx.x;
    if (i < n) p[i] = 0;
}

__global__ __launch_bounds__(256) void k_count(const int* __restrict__ dst, int* __restrict__ cnt, int e) {
    int i = blockIdx.x * blockDim.x + threadIdx.x;
    if (i < e) atomicAdd(&cnt[dst[i]], 1);
}

__global__ __launch_bounds__(256) void k_dinv(const int* __restrict__ cnt, float* __restrict__ dinv, int n) {
    int i = blockIdx.x * blockDim.x + threadIdx.x;
    if (i < n) dinv[i] = rsqrtf(1.0f + (float)cnt[i]);   // +1 self loop; always > 0
}

// single-workgroup exclusive scan (runs once; N = 100k -> ~98 chunks of 1024)
__global__ __launch_bounds__(1024) void k_scan(const int* __restrict__ cnt, int* __restrict__ rowptr, int n) {
    __shared__ int sm[1024];
    __shared__ int carry_sm;
    int tid = threadIdx.x;
    if (tid == 0) carry_sm = 0;
    __syncthreads();
    for (int base = 0; base < n; base += 1024) {
        int i = base + tid;
        int v = (i < n) ? cnt[i] : 0;
        sm[tid] = v;
        __syncthreads();
        #pragma unroll
        for (int off = 1; off < 1024; off <<= 1) {
            int t = (tid >= off) ? sm[tid - off] : 0;
            __syncthreads();
            sm[tid] += t;
            __syncthreads();
        }
        int carry = carry_sm;
        if (i < n) rowptr[i] = carry + sm[tid] - v;   // exclusive
        __syncthreads();
        if (tid == 0) carry_sm = carry + sm[1023];
        __syncthreads();
    }
    if (tid == 0) rowptr[n] = carry_sm;
}

__global__ __launch_bounds__(256) void k_copy_int(const int* __restrict__ a, int* __restrict__ b, int n) {
    int i = blockIdx.x * blockDim.x + threadIdx.x;
    if (i < n) b[i] = a[i];
}

__global__ __launch_bounds__(256) void k_fill(const int* __restrict__ src, const int* __restrict__ dst,
                                              int* __restrict__ pos, int* __restrict__ col, int e) {
    int i = blockIdx.x * blockDim.x + threadIdx.x;
    if (i < e) {
        int p = atomicAdd(&pos[dst[i]], 1);
        col[p] = src[i];
    }
}

// transpose W (KDIM x DOUT, row-major) -> Wt (DOUT x KDIM, row-major)
__global__ __launch_bounds__(256) void k_transpose(const float* __restrict__ W, float* __restrict__ Wt,
                                                   int dout, int total) {
    int i = blockIdx.x * blockDim.x + threadIdx.x;
    if (i < total) {
        int k = i / dout;
        int c = i - k * dout;
        Wt[(size_t)c * KDIM + k] = W[i];
    }
}

// ---------------------------------------------------------------- GEMM: G = dinv .* (H @ W)
// K = 128. One wave computes 32 rows (2 row-tiles) x DOUT cols using fp32 WMMA 16x16x4.
// Wt is the transposed weight (DOUT x 128): B fragment = one b64 load per tile.
template<int DOUT>
__global__ __launch_bounds__(256) void k_gemm(const float* __restrict__ H, const float* __restrict__ Wt,
                                              const float* __restrict__ dinv, float* __restrict__ G, int n) {
    constexpr int NT = DOUT / 16;
    const int lane = threadIdx.x & 31;
    const int wave = threadIdx.x >> 5;
    const int row0 = (blockIdx.x * 8 + wave) * 32;   // 8 waves/block, 32 rows/wave
    if (row0 >= n) return;

    const int mrow = lane & 15;           // M (or N) index within 16-wide tile
    const int kgrp = (lane >> 4) << 1;    // lanes 16-31 hold K+2,K+3

    v8f acc0[NT], acc1[NT];
    #pragma unroll
    for (int t = 0; t < NT; t++) { acc0[t] = (v8f)0.0f; acc1[t] = (v8f)0.0f; }

    // clamp rows (branchless loads; out-of-range rows never stored)
    int r0 = row0 + mrow;      if (r0 > n - 1) r0 = n - 1;
    int r1 = row0 + 16 + mrow; if (r1 > n - 1) r1 = n - 1;
    const float* hp0 = H + (size_t)r0 * KDIM + kgrp;
    const float* hp1 = H + (size_t)r1 * KDIM + kgrp;

    for (int k = 0; k < KDIM; k += 4) {
        v2f a0 = *(const v2f*)(hp0 + k);
        v2f a1 = *(const v2f*)(hp1 + k);
        #pragma unroll
        for (int t = 0; t < NT; t++) {
            // B layout: VGPR0 = W[k+kgrp][col], VGPR1 = W[k+kgrp+1][col]; contiguous in Wt
            v2f b = *(const v2f*)(Wt + (size_t)(t * 16 + mrow) * KDIM + k + kgrp);
            acc0[t] = __builtin_amdgcn_wmma_f32_16x16x4_f32(false, a0, false, b, (short)0, acc0[t], false, false);
            acc1[t] = __builtin_amdgcn_wmma_f32_16x16x4_f32(false, a1, false, b, (short)0, acc1[t], false, false);
        }
    }

    // epilogue: D layout — VGPR r: lanes0-15 -> M=r, lanes16-31 -> M=r+8
    const int hi8 = (lane >> 4) << 3;
    #pragma unroll
    for (int r = 0; r < 8; r++) {
        int rowA = row0 + r + hi8;
        int rowB = rowA + 16;
        float dvA = (rowA < n) ? dinv[rowA] : 0.0f;
        float dvB = (rowB < n) ? dinv[rowB] : 0.0f;
        #pragma unroll
        for (int t = 0; t < NT; t++) {
            int c = t * 16 + (lane & 15);
            if (rowA < n) G[(size_t)rowA * DOUT + c] = dvA * acc0[t][r];
            if (rowB < n) G[(size_t)rowB * DOUT + c] = dvB * acc1[t][r];
        }
    }
}

// ---------------------------------------------------------------- fused aggregation
// out[i] = act( dinv[i] * (g[i] + sum_{e in CSR(i)} g[col[e]]) + b )
template<int DOUT>
struct VecSel;
template<> struct VecSel<128> { typedef __attribute__((ext_vector_type(4))) float T; static const int V = 4; };
template<> struct VecSel<64>  { typedef __attribute__((ext_vector_type(2))) float T; static const int V = 2; };

template<int DOUT, bool RELU>
__global__ __launch_bounds__(256) void k_agg(const float* __restrict__ G, const int* __restrict__ rowptr,
                                             const int* __restrict__ col, const float* __restrict__ dinv,
                                             const float* __restrict__ bias, float* __restrict__ OUT, int n) {
    typedef typename VecSel<DOUT>::T vec;
    constexpr int V = VecSel<DOUT>::V;
    int node = (blockIdx.x * blockDim.x + threadIdx.x) >> 5;  // one wave per node
    int lane = threadIdx.x & 31;
    if (node >= n) return;

    int beg = rowptr[node];
    int end = rowptr[node + 1];

    vec s = *(const vec*)(G + (size_t)node * DOUT + lane * V);     // self loop
    int e = beg;
    // 4-way unroll: 4 index loads + 4 row loads in flight per iteration
    for (; e + 3 < end; e += 4) {
        int s0 = col[e], s1 = col[e + 1], s2 = col[e + 2], s3 = col[e + 3];
        vec a0 = *(const vec*)(G + (size_t)s0 * DOUT + lane * V);
        vec a1 = *(const vec*)(G + (size_t)s1 * DOUT + lane * V);
        vec a2 = *(const vec*)(G + (size_t)s2 * DOUT + lane * V);
        vec a3 = *(const vec*)(G + (size_t)s3 * DOUT + lane * V);
        s += a0; s += a1; s += a2; s += a3;
    }
    for (; e < end; e++) {
        int sc = col[e];
        s += *(const vec*)(G + (size_t)sc * DOUT + lane * V);
    }
    float dv = dinv[node];
    vec o;
    #pragma unroll
    for (int j = 0; j < V; j++) {
        float x = dv * s[j] + bias[lane * V + j];
        if (RELU) x = fmaxf(x, 0.0f);
        o[j] = x;
    }
    *(vec*)(OUT + (size_t)node * DOUT + lane * V) = o;
}

// ---------------------------------------------------------------- launcher
extern "C" void kernel_launch(void* const* d_in, const int* in_sizes, int n_in,
                              void* d_out, int out_size, void* d_ws, size_t ws_size,
                              hipStream_t stream) {
    const float* x  = (const float*)d_in[0];
    const int*   ei = (const int*)d_in[1];
    const float* W0 = (const float*)d_in[2];
    const float* b0 = (const float*)d_in[3];
    const float* W1 = (const float*)d_in[4];
    const float* b1 = (const float*)d_in[5];
    const float* W2 = (const float*)d_in[6];
    const float* b2 = (const float*)d_in[7];

    const int N = in_sizes[0] / 128;
    const int E = in_sizes[1] / 2;
    const int* src = ei;
    const int* dst = ei + E;

    // workspace carve-out (256B aligned)
    char* ws = (char*)d_ws;
    auto carve = [&](size_t bytes) -> void* {
        void* p = (void*)ws;
        ws += (bytes + 255) & ~(size_t)255;
        return p;
    };
    float* dinv   = (float*)carve((size_t)N * 4);
    int*   cnt    = (int*)  carve((size_t)N * 4);
    int*   rowptr = (int*)  carve((size_t)(N + 1) * 4);
    int*   pos    = (int*)  carve((size_t)N * 4);
    int*   col    = (int*)  carve((size_t)E * 4);
    float* wt     = (float*)carve((size_t)128 * 128 * 4);   // transposed weights (reused)
    float* g      = (float*)carve((size_t)N * 128 * 4);
    float* h      = (float*)carve((size_t)N * 128 * 4);

    const int TB = 256;
    int gN = (N + TB - 1) / TB;
    int gE = (E + TB - 1) / TB;

    // CSR build + normalization (runs every call; deterministic work)
    k_zero_int<<<gN, TB, 0, stream>>>(cnt, N);
    k_count<<<gE, TB, 0, stream>>>(dst, cnt, E);
    k_dinv<<<gN, TB, 0, stream>>>(cnt, dinv, N);
    k_scan<<<1, 1024, 0, stream>>>(cnt, rowptr, N);
    k_copy_int<<<gN, TB, 0, stream>>>(rowptr, pos, N);
    k_fill<<<gE, TB, 0, stream>>>(src, dst, pos, col, E);

    int gGemm = (N + 255) / 256;          // 256 rows per block (8 waves x 32 rows)
    int gAgg  = (N + 7) / 8;              // 8 waves (nodes) per block
    int gT128 = (128 * 128 + TB - 1) / TB;
    int gT64  = (128 * 64 + TB - 1) / TB;

    // layer 0: x -> h
    k_transpose<<<gT128, TB, 0, stream>>>(W0, wt, 128, 128 * 128);
    k_gemm<128><<<gGemm, TB, 0, stream>>>(x, wt, dinv, g, N);
    k_agg<128, true><<<gAgg, TB, 0, stream>>>(g, rowptr, col, dinv, b0, h, N);
    // layer 1: h -> h
    k_transpose<<<gT128, TB, 0, stream>>>(W1, wt, 128, 128 * 128);
    k_gemm<128><<<gGemm, TB, 0, stream>>>(h, wt, dinv, g, N);
    k_agg<128, true><<<gAgg, TB, 0, stream>>>(g, rowptr, col, dinv, b1, h, N);
    // layer 2: h -> out (no relu)
    k_transpose<<<gT64, TB, 0, stream>>>(W2, wt, 64, 128 * 64);
    k_gemm<64><<<gGemm, TB, 0, stream>>>(h, wt, dinv, g, N);
    k_agg<64, false><<<gAgg, TB, 0, stream>>>(g, rowptr, col, dinv, b2, (float*)d_out, N);
}